// Top1Gate_66614942761618
// MI455X (gfx1250) — compile-verified
//
#include <hip/hip_runtime.h>
#include <hip/hip_bf16.h>

// Problem constants (from reference)
#define S_TOKENS 8192
#define DMODEL   2048
#define NEXPERT  64
#define CAP      128                       // capacity = ceil(S/E) * 1.0
#define TPB_TOK  64                        // tokens handled per gate block
#define NBLK     (S_TOKENS / TPB_TOK)      // 128 gate blocks

typedef float v2f __attribute__((ext_vector_type(2)));
typedef float v8f __attribute__((ext_vector_type(8)));

// ---------------------------------------------------------------------------
// Kernel 1: gate GEMM (f32 WMMA 16x16x4) + softmax + argmax + per-block
//           ranks/histogram/me partials.  One wave computes a 16-token x
//           64-expert logits tile; workgroup = 4 waves = 64 tokens.
// ---------------------------------------------------------------------------
__global__ __launch_bounds__(128)
void gate_gemm_softmax_kernel(const float* __restrict__ x,    // [S, D]
                              const float* __restrict__ wg,   // [E, D]
                              int*   __restrict__ expertIdx,  // [S]
                              float* __restrict__ gate1,      // [S]
                              int*   __restrict__ rankInBlk,  // [S]
                              int*   __restrict__ blockCounts,// [NBLK][E]
                              float* __restrict__ meBlock)    // [NBLK][E]
{
    __shared__ float llog[TPB_TOK * (NEXPERT + 1)];  // 64 x 65 (pad -> no bank conflicts)
    __shared__ float maxv[TPB_TOK];
    __shared__ float sumv[TPB_TOK];
    __shared__ int   eidx[TPB_TOK];

    const int tid  = threadIdx.x;
    const int wave = tid >> 5;
    const int lane = tid & 31;
    const int half = lane >> 4;      // lanes 16..31 carry K+2 (A) / row+2 (B)
    const int m    = lane & 15;      // M for A, N for B/C/D
    const int blk  = blockIdx.x;
    const int tokBase = blk * TPB_TOK;
    const int rowA = tokBase + wave * 16 + m;

    // --- WMMA GEMM: logits[16 tok][64 exp] = x_tile (16xK) * wg^T (Kx64) ---
    v8f acc[4] = {};   // expert tiles e0 = 0,16,32,48
    const float* xrow = x + (size_t)rowA * DMODEL;

    #pragma unroll 4
    for (int kb = 0; kb < DMODEL; kb += 4) {
        // A 16x4 f32: VGPR v holds K = 2*half + v  -> contiguous pair
        v2f a = *(const v2f*)(xrow + kb + 2 * half);
        #pragma unroll
        for (int t = 0; t < 4; ++t) {
            // B 4x16 f32: VGPR v holds K-row = 2*half + v, N = m
            const float* wrow = wg + (size_t)(t * 16 + m) * DMODEL + kb + 2 * half;
            v2f b = *(const v2f*)wrow;
#if __has_builtin(__builtin_amdgcn_wmma_f32_16x16x4_f32)
            acc[t] = __builtin_amdgcn_wmma_f32_16x16x4_f32(
                false, a, false, b, (short)0, acc[t], false, false);
#else
            // compile-safe placeholder (builtin expected to exist on gfx1250)
            acc[t][0] += a.x * b.x;
#endif
        }
    }

    // C/D 16x16: VGPR v at lane(half,m) holds [M = v + 8*half][N = m]
    #pragma unroll
    for (int t = 0; t < 4; ++t) {
        #pragma unroll
        for (int v = 0; v < 8; ++v) {
            int tokLocal = wave * 16 + v + 8 * half;
            int e        = t * 16 + m;
            llog[tokLocal * (NEXPERT + 1) + e] = acc[t][v];
        }
    }
    __syncthreads();

    // --- per-token softmax stats + top-1 (first-max tie-break like argmax) ---
    if (tid < TPB_TOK) {
        const float* row = &llog[tid * (NEXPERT + 1)];
        float mx = row[0];
        int   bi = 0;
        for (int e = 1; e < NEXPERT; ++e) {
            float v = row[e];
            if (v > mx) { mx = v; bi = e; }
        }
        float s = 0.0f;
        for (int e = 0; e < NEXPERT; ++e) s += __expf(row[e] - mx);
        maxv[tid] = mx;
        sumv[tid] = s;
        eidx[tid] = bi;
        int gs = tokBase + tid;
        expertIdx[gs] = bi;
        gate1[gs]     = 1.0f / s;   // softmax value at the argmax = exp(0)/sum
    }
    __syncthreads();

    if (tid < TPB_TOK) {
        // rank of token within its block among same-expert tokens (in order)
        int e = eidx[tid];
        int r = 0;
        for (int t2 = 0; t2 < tid; ++t2) r += (eidx[t2] == e) ? 1 : 0;
        rankInBlk[tokBase + tid] = r;
    } else if (tid < 2 * TPB_TOK) {
        // per-block expert histogram + deterministic me partial sums
        int e = tid - TPB_TOK;
        int cnt = 0;
        float ms = 0.0f;
        for (int t2 = 0; t2 < TPB_TOK; ++t2) {
            cnt += (eidx[t2] == e) ? 1 : 0;
            ms  += __expf(llog[t2 * (NEXPERT + 1) + e] - maxv[t2]) / sumv[t2];
        }
        blockCounts[blk * NEXPERT + e] = cnt;
        meBlock[blk * NEXPERT + e]     = ms;
    }
}

// ---------------------------------------------------------------------------
// Kernel 2: per-expert exclusive scan over blocks + aux loss (deterministic).
// ---------------------------------------------------------------------------
__global__ __launch_bounds__(64)
void expert_scan_kernel(const int*   __restrict__ blockCounts,
                        const float* __restrict__ meBlock,
                        int*   __restrict__ blockBase,
                        float* __restrict__ l_aux_out)
{
    __shared__ float red[NEXPERT];
    int e = threadIdx.x;
    int base = 0;
    float me = 0.0f;
    for (int b = 0; b < NBLK; ++b) {
        blockBase[b * NEXPERT + e] = base;
        base += blockCounts[b * NEXPERT + e];
        me   += meBlock[b * NEXPERT + e];
    }
    float mee = me   / (float)S_TOKENS;              // mean gate prob
    float ce  = (float)base / (float)S_TOKENS;       // selection fraction
    red[e] = mee * ce;
    __syncthreads();
    for (int off = 32; off > 0; off >>= 1) {
        if (e < off) red[e] += red[e + off];
        __syncthreads();
    }
    // l_aux = mean_e(me*ce) * E*E = sum_e(me*ce) * E
    if (e == 0) l_aux_out[0] = red[0] * (float)NEXPERT;
}

// ---------------------------------------------------------------------------
// Kernel 3: stream-zero the 512MB output region (HBM-write-bound, float4).
// ---------------------------------------------------------------------------
__global__ __launch_bounds__(256)
void zero_out_kernel(float4* __restrict__ p, size_t nvec, float* __restrict__ tail)
{
    size_t i      = (size_t)blockIdx.x * blockDim.x + threadIdx.x;
    size_t stride = (size_t)gridDim.x * blockDim.x;
    float4 z = make_float4(0.f, 0.f, 0.f, 0.f);
    for (; i < nvec; i += stride) p[i] = z;
    if (blockIdx.x == 0 && threadIdx.x == 0) *tail = 0.f;
}

// ---------------------------------------------------------------------------
// Kernel 4: scatter the 8192 nonzeros into combine / dispatch mask.
// ---------------------------------------------------------------------------
__global__ __launch_bounds__(256)
void scatter_kernel(const int*   __restrict__ expertIdx,
                    const float* __restrict__ gate1,
                    const int*   __restrict__ rankInBlk,
                    const int*   __restrict__ blockBase,
                    float* __restrict__ combine,
                    float* __restrict__ maskf)
{
    int s = blockIdx.x * blockDim.x + threadIdx.x;
    if (s >= S_TOKENS) return;
    int e   = expertIdx[s];
    int blk = s / TPB_TOK;
    int loc = blockBase[blk * NEXPERT + e] + rankInBlk[s];
    if (loc < CAP) {   // capacity drop: tokens beyond capacity leave zero rows
        size_t off = ((size_t)s * NEXPERT + e) * CAP + (size_t)loc;
        combine[off] = gate1[s];
        maskf[off]   = 1.0f;
    }
}

// ---------------------------------------------------------------------------
extern "C" void kernel_launch(void* const* d_in, const int* in_sizes, int n_in,
                              void* d_out, int out_size, void* d_ws, size_t ws_size,
                              hipStream_t stream)
{
    const float* x  = (const float*)d_in[0];   // [S, D]
    const float* wg = (const float*)d_in[1];   // [E, D]
    float* out = (float*)d_out;

    // workspace carve (~192 KB)
    char* ws = (char*)d_ws;
    int*   expertIdx   = (int*)ws;   ws += (size_t)S_TOKENS * 4;
    int*   rankInBlk   = (int*)ws;   ws += (size_t)S_TOKENS * 4;
    float* gate1       = (float*)ws; ws += (size_t)S_TOKENS * 4;
    int*   blockCounts = (int*)ws;   ws += (size_t)NBLK * NEXPERT * 4;
    int*   blockBase   = (int*)ws;   ws += (size_t)NBLK * NEXPERT * 4;
    float* meBlock     = (float*)ws; ws += (size_t)NBLK * NEXPERT * 4;

    const size_t big = (size_t)S_TOKENS * NEXPERT * CAP;  // 67,108,864
    float* l_aux   = out;            // output order: l_aux, combine, dispatch
    float* combine = out + 1;
    float* maskf   = out + 1 + big;

    // 1) gate GEMM + softmax + per-block stats
    gate_gemm_softmax_kernel<<<NBLK, 128, 0, stream>>>(
        x, wg, expertIdx, gate1, rankInBlk, blockCounts, meBlock);

    // 2) zero the full output (2*big + 1 floats); tail handles the odd element
    size_t totalFloats = 2 * big + 1;
    size_t nvec = totalFloats / 4;
    zero_out_kernel<<<2048, 256, 0, stream>>>((float4*)out, nvec, out + nvec * 4);

    // 3) cross-block scan + aux loss (writes l_aux after the zero pass)
    expert_scan_kernel<<<1, 64, 0, stream>>>(blockCounts, meBlock, blockBase, l_aux);

    // 4) scatter nonzeros
    scatter_kernel<<<(S_TOKENS + 255) / 256, 256, 0, stream>>>(
        expertIdx, gate1, rankInBlk, blockBase, combine, maskf);
}